// tinyLayerE_18726057410768
// MI455X (gfx1250) — compile-verified
//
#include <hip/hip_runtime.h>
#include <hip/hip_bf16.h>
#include <math.h>

typedef float v2f __attribute__((ext_vector_type(2)));
typedef float v8f __attribute__((ext_vector_type(8)));

#define NFEAT 20000
#define KDIM  64
#define DDIM  256
#define BATCH 4096

// ---------------------------------------------------------------------------
// Kernel 1: Z = u (20000x64) @ tinyW (64x256), fp32 WMMA 16x16x4.
// One wave (32 threads) per 16x16 output tile; 16 WMMA issues over K=64.
// A layout (16x4 f32): lanes 0-15 -> M=lane, {K=0,K=1}; lanes 16-31 -> {K=2,K=3}
// B layout (4x16 f32): lanes 0-15 -> N=lane, rows K=0,1; lanes 16-31 -> K=2,3
// C layout (16x16 f32): VGPR v -> M = v + 8*(lane>=16), N = lane&15
// ---------------------------------------------------------------------------
__global__ void gemm_wmma_f32_kernel(const float* __restrict__ A,
                                     const float* __restrict__ B,
                                     float* __restrict__ Z) {
  int tile  = blockIdx.x;
  int mTile = tile >> 4;      // 0..1249
  int nTile = tile & 15;      // 0..15
  int lane  = threadIdx.x;    // 0..31 (wave32)
  int half  = lane >> 4;      // 0 or 1
  int l     = lane & 15;

  v8f c = {};
  const float* arow = A + (size_t)(mTile * 16 + l) * KDIM;
  const float* bbase = B + (size_t)(nTile * 16 + l);
  for (int k0 = 0; k0 < KDIM; k0 += 4) {
    int ka = k0 + half * 2;
    v2f a, b;
    a.x = arow[ka];
    a.y = arow[ka + 1];
    b.x = bbase[(size_t)ka * DDIM];
    b.y = bbase[(size_t)(ka + 1) * DDIM];
    c = __builtin_amdgcn_wmma_f32_16x16x4_f32(
        /*neg_a=*/false, a, /*neg_b=*/false, b,
        /*c_mod=*/(short)0, c, /*reuse_a=*/false, /*reuse_b=*/false);
  }

  int n = nTile * 16 + l;
  int mBase = mTile * 16 + half * 8;
  #pragma unroll
  for (int v = 0; v < 8; ++v) {
    Z[(size_t)(mBase + v) * DDIM + n] = c[v];
  }
}

// ---------------------------------------------------------------------------
// Kernel 2: per-n softmax over D (column softmax of samples), fused:
//   logp   = z - max - log(sum exp)
//   logits = log(10) + clamp(logp, log(1e-7), log(0.9999999))
//   g      = -log(-log(uniform))
//   T[d,n] = (logits + g) / 9.9999
// One thread per n; writes coalesced along n.
// ---------------------------------------------------------------------------
__global__ void col_softmax_logits_kernel(const float* __restrict__ Z,
                                          const float* __restrict__ uniform,
                                          float* __restrict__ T) {
  int n = blockIdx.x * blockDim.x + threadIdx.x;
  if (n >= NFEAT) return;
  const float* zrow = Z + (size_t)n * DDIM;

  float m = -INFINITY;
  for (int d = 0; d < DDIM; ++d) m = fmaxf(m, zrow[d]);
  float s = 0.0f;
  for (int d = 0; d < DDIM; ++d) s += __expf(zrow[d] - m);
  float logS = __logf(s);

  const float LOG10   = 2.302585092994046f;
  const float LO      = -16.11809565095832f;   // log(1e-7)
  const float HI      = -1.00000005e-7f;       // log(0.9999999)
  const float INVTEMP = 1.0f / 9.9999f;        // max(0.1, 10*0.99999)

  for (int d = 0; d < DDIM; ++d) {
    float logp = zrow[d] - m - logS;
    logp = fminf(fmaxf(logp, LO), HI);
    float uu = uniform[(size_t)d * NFEAT + n];
    float g = -__logf(-__logf(uu));
    T[(size_t)d * NFEAT + n] = (LOG10 + logp + g) * INVTEMP;
  }
}

// ---------------------------------------------------------------------------
// Kernel 3: per-row (D rows) softmax over N=20000, normalize T in place.
// ---------------------------------------------------------------------------
__global__ void row_softmax_kernel(float* __restrict__ T) {
  __shared__ float red[512];
  int d = blockIdx.x;
  int t = threadIdx.x;
  float* row = T + (size_t)d * NFEAT;

  float m = -INFINITY;
  for (int n = t; n < NFEAT; n += blockDim.x) m = fmaxf(m, row[n]);
  red[t] = m;
  __syncthreads();
  for (int s = blockDim.x / 2; s > 0; s >>= 1) {
    if (t < s) red[t] = fmaxf(red[t], red[t + s]);
    __syncthreads();
  }
  m = red[0];
  __syncthreads();

  float sum = 0.0f;
  for (int n = t; n < NFEAT; n += blockDim.x) sum += __expf(row[n] - m);
  red[t] = sum;
  __syncthreads();
  for (int s = blockDim.x / 2; s > 0; s >>= 1) {
    if (t < s) red[t] += red[t + s];
    __syncthreads();
  }
  float invS = 1.0f / red[0];
  __syncthreads();

  for (int n = t; n < NFEAT; n += blockDim.x)
    row[n] = __expf(row[n] - m) * invS;
}

// ---------------------------------------------------------------------------
// Kernel 4: 256-step greedy global argmax with row/column suppression.
// Single 1024-thread block. Per-row cached argmax over non-taken columns;
// a row is re-scanned only when its cached argmax column gets taken.
// ---------------------------------------------------------------------------
__global__ void __launch_bounds__(1024)
greedy_select_kernel(const float* __restrict__ S, int* __restrict__ colIdx) {
  __shared__ float rowBest[DDIM];
  __shared__ int   rowBestCol[DDIM];
  __shared__ unsigned char rowTaken[DDIM];
  __shared__ unsigned char need[DDIM];
  __shared__ unsigned colTaken[(NFEAT + 31) / 32];  // 625 words
  __shared__ float rv[1024];
  __shared__ int   ri[1024];
  __shared__ int   selY, recRow;

  int t = threadIdx.x;
  for (int i = t; i < (NFEAT + 31) / 32; i += 1024) colTaken[i] = 0u;
  if (t < DDIM) { rowTaken[t] = 0; need[t] = 0; }
  __syncthreads();

  // Initial per-row argmax: 4 threads per row, 5000 elements each.
  {
    int d = t >> 2, q = t & 3;
    int chunk = (NFEAT + 3) / 4;
    int n0 = q * chunk;
    int n1 = n0 + chunk; if (n1 > NFEAT) n1 = NFEAT;
    const float* row = S + (size_t)d * NFEAT;
    float bv = -INFINITY; int bi = 0;
    for (int n = n0; n < n1; ++n) {
      float v = row[n];
      if (v > bv) { bv = v; bi = n; }
    }
    rv[t] = bv; ri[t] = bi;
  }
  __syncthreads();
  if ((t & 3) == 0) {
    float bv = rv[t]; int bi = ri[t];
    for (int q = 1; q < 4; ++q)
      if (rv[t + q] > bv) { bv = rv[t + q]; bi = ri[t + q]; }
    rowBest[t >> 2] = bv; rowBestCol[t >> 2] = bi;
  }
  __syncthreads();

  for (int it = 0; it < DDIM; ++it) {
    // Reduce 256 row maxima -> global (x, y).
    if (t < DDIM) {
      rv[t] = rowTaken[t] ? -INFINITY : rowBest[t];
      ri[t] = t;
    }
    __syncthreads();
    for (int s = DDIM / 2; s > 0; s >>= 1) {
      if (t < s && rv[t + s] > rv[t]) { rv[t] = rv[t + s]; ri[t] = ri[t + s]; }
      __syncthreads();
    }
    if (t == 0) {
      int x = ri[0];
      int y = rowBestCol[x];
      selY = y;
      rowTaken[x] = 1;
      colTaken[y >> 5] |= (1u << (y & 31));
      colIdx[x] = y;
    }
    __syncthreads();
    int y = selY;
    if (t < DDIM) need[t] = (!rowTaken[t] && rowBestCol[t] == y) ? 1 : 0;
    __syncthreads();

    // Re-scan any live row whose cached argmax column was just taken.
    while (true) {
      if (t == 0) {
        int r = -1;
        for (int d = 0; d < DDIM; ++d) if (need[d]) { r = d; break; }
        recRow = r;
      }
      __syncthreads();
      int r = recRow;
      if (r < 0) break;
      const float* row = S + (size_t)r * NFEAT;
      float bv = -INFINITY; int bi = 0;
      for (int n = t; n < NFEAT; n += 1024) {
        if (colTaken[n >> 5] & (1u << (n & 31))) continue;
        float v = row[n];
        if (v > bv) { bv = v; bi = n; }
      }
      rv[t] = bv; ri[t] = bi;
      __syncthreads();
      for (int s = 512; s > 0; s >>= 1) {
        if (t < s && rv[t + s] > rv[t]) { rv[t] = rv[t + s]; ri[t] = ri[t + s]; }
        __syncthreads();
      }
      if (t == 0) { rowBest[r] = rv[0]; rowBestCol[r] = ri[0]; need[r] = 0; }
      __syncthreads();
    }
  }
}

// ---------------------------------------------------------------------------
// Kernel 5: Y = X @ dl.T with one-hot dl == column gather.
// ---------------------------------------------------------------------------
__global__ void gather_kernel(const float* __restrict__ X,
                              const int* __restrict__ colIdx,
                              float* __restrict__ Y) {
  int d = threadIdx.x;   // 0..255
  int b = blockIdx.x;    // 0..4095
  Y[(size_t)b * DDIM + d] = X[(size_t)b * NFEAT + colIdx[d]];
}

// ---------------------------------------------------------------------------
extern "C" void kernel_launch(void* const* d_in, const int* in_sizes, int n_in,
                              void* d_out, int out_size, void* d_ws, size_t ws_size,
                              hipStream_t stream) {
  const float* X       = (const float*)d_in[0];  // 4096 x 20000
  const float* u       = (const float*)d_in[1];  // 20000 x 64
  const float* tinyW   = (const float*)d_in[2];  // 64 x 256
  const float* uniform = (const float*)d_in[3];  // 256 x 20000
  float* out = (float*)d_out;                    // 4096 x 256

  char* ws = (char*)d_ws;
  const size_t matBytes = (size_t)NFEAT * DDIM * sizeof(float);  // 20.48 MB
  float* Z = (float*)ws;                          // 20000 x 256
  float* T = (float*)(ws + matBytes);             // 256 x 20000 (pre/post softmax)
  int* colIdx = (int*)(ws + 2 * matBytes);        // 256 ints

  gemm_wmma_f32_kernel<<<(NFEAT / 16) * (DDIM / 16), 32, 0, stream>>>(u, tinyW, Z);
  col_softmax_logits_kernel<<<(NFEAT + 255) / 256, 256, 0, stream>>>(Z, uniform, T);
  row_softmax_kernel<<<DDIM, 512, 0, stream>>>(T);
  greedy_select_kernel<<<1, 1024, 0, stream>>>(T, colIdx);
  gather_kernel<<<BATCH, DDIM, 0, stream>>>(X, colIdx, out);
}